// PlasticCritic_40638980555359
// MI455X (gfx1250) — compile-verified
//
#include <hip/hip_runtime.h>

#define A_   8
#define B_N  32768
#define IN_  128
#define OUT_ 32
#define H_   128
#define KH   4
#define DH   32
#define SAF  160           // IN_ + OUT_
#define BT   64            // batch rows per GEMM tile
#define ATT_BT 8
#define EPS_ 1e-5f

typedef float v2f __attribute__((ext_vector_type(2)));
typedef float v8f __attribute__((ext_vector_type(8)));
typedef unsigned int v4u __attribute__((ext_vector_type(4)));
typedef int v4i __attribute__((ext_vector_type(4)));
typedef int v8i __attribute__((ext_vector_type(8)));

__device__ __forceinline__ v8f wmma4(v2f a, v2f b, v8f c) {
  return __builtin_amdgcn_wmma_f32_16x16x4_f32(false, a, false, b, (short)0, c, false, false);
}

__device__ __forceinline__ float lrelu(float x) { return x > 0.f ? x : 0.01f * x; }

__device__ __forceinline__ void zero4(v8f acc[4]) {
  v8f z = {0.f, 0.f, 0.f, 0.f, 0.f, 0.f, 0.f, 0.f};
#pragma unroll
  for (int t = 0; t < 4; ++t) acc[t] = z;
}

// ---------------- Tensor Data Mover: tiled global -> LDS with 1-dword pad per 32 dwords
// Builds a D# per CDNA5 ISA ch.8: group0 = {count/lds/global/type}, group1 = {flags,dims,strides}.
// 2D tile: tile_d2 == 0. 3D tile (attention): tile_d2 = slices, stride1 = dim1_stride (elements).
__device__ __forceinline__ void tdm_load(unsigned lds_off, const void* gptr,
                                         unsigned tensor_d0, unsigned tensor_d1,
                                         unsigned tile_d0, unsigned tile_d1, unsigned tile_d2,
                                         unsigned long long stride0, unsigned long long stride1,
                                         unsigned tensor_d2) {
  const unsigned long long ga = (unsigned long long)(uintptr_t)gptr;
  v4u g0;
  g0.x = 1u;                                                  // count=1, user descriptor
  g0.y = lds_off;                                             // lds_addr (bytes)
  g0.z = (unsigned)(ga & 0xffffffffull);                      // global_addr[31:0]
  g0.w = (unsigned)((ga >> 32) & 0x1ffffffull) | (2u << 30);  // global_addr[56:32], type=2
  v8i g1;
  g1[0] = (int)((2u << 16) | (1u << 20) | (4u << 22));        // data_size=4B, pad_en, 1dw per 32dw
  g1[1] = (int)((tensor_d0 & 0xffffu) << 16);                 // tensor_dim0[15:0]
  g1[2] = (int)((tensor_d0 >> 16) | ((tensor_d1 & 0xffffu) << 16));
  g1[3] = (int)((tensor_d1 >> 16) | (tile_d0 << 16));
  g1[4] = (int)(tile_d1 | (tile_d2 << 16));
  g1[5] = (int)(stride0 & 0xffffffffull);
  g1[6] = (int)((unsigned)(stride0 >> 32) | (((unsigned)stride1 & 0xffffu) << 16));
  g1[7] = (int)(stride1 >> 16);
  v4i g2; g2[0] = (int)tensor_d2; g2[1] = 0; g2[2] = 0; g2[3] = 0;
  v4i g3 = {0, 0, 0, 0};
#if __clang_major__ >= 23
  v8i g4 = {0, 0, 0, 0, 0, 0, 0, 0};
  __builtin_amdgcn_tensor_load_to_lds(g0, g1, g2, g3, g4, 0);
#else
  __builtin_amdgcn_tensor_load_to_lds(g0, g1, g2, g3, 0);
#endif
}

// TDM-padded LDS row index: element f of a row stored at f + f/32 (row stride = len + len/32)
__device__ __forceinline__ int padidx(int f) { return f + (f >> 5); }

// ---------------- WMMA GEMM: wave owns one 16-col strip, 4 row blocks (64 rows) ----------
// WT layout: [outcol][kk] contiguous (pre-transposed), so a B fragment is one b64 load.
template <typename XF>
__device__ __forceinline__ void gemm_col(XF&& xf, int kdim, const float* __restrict__ WT,
                                         int ldw, int wv, int m, int kh, v8f acc[4]) {
  const float* wrow = WT + (size_t)(wv * 16 + m) * ldw;
  for (int kk0 = 0; kk0 < kdim; kk0 += 4) {
    const int kka = kk0 + kh * 2;                 // lane-half selects K pair (ISA 7.12.2)
    const v2f bfr = *(const v2f*)(wrow + kka);    // aligned: ldw even, kka even
#pragma unroll
    for (int rb = 0; rb < 4; ++rb) {
      v2f afr;
      afr.x = xf(rb * 16 + m, kka);
      afr.y = xf(rb * 16 + m, kka + 1);
      acc[rb] = wmma4(afr, bfr, acc[rb]);
    }
  }
}

__device__ __forceinline__ void store_col(v8f acc[4], const float* __restrict__ bias, bool act,
                                          float* __restrict__ gout, size_t grow0,
                                          float* __restrict__ lout, int ldl,
                                          int wv, int m, int kh) {
  const int col = wv * 16 + m;
  const float bv = bias ? bias[col] : 0.f;
#pragma unroll
  for (int rb = 0; rb < 4; ++rb)
#pragma unroll
    for (int v = 0; v < 8; ++v) {
      const int row = rb * 16 + v + 8 * kh;       // C/D: VGPR v -> M = v (+8 for hi lanes)
      float x = acc[rb][v] + bv;
      if (act) x = lrelu(x);
      if (gout) gout[(grow0 + (size_t)row) * H_ + col] = x;
      if (lout) lout[row * ldl + col] = x;
    }
}

// ---------------- weight transposes (tiny, amortized) ----------------

__global__ void transpose_std_kernel(const float* __restrict__ W, float* __restrict__ WT,
                                     int nmat, int kdim) {
  // W: [nmat][kdim][H_]  ->  WT: [nmat][H_][kdim]
  const int i = blockIdx.x * 256 + threadIdx.x;
  if (i >= nmat * kdim * H_) return;
  const int c = i % H_, kk = (i / H_) % kdim, mat = i / (H_ * kdim);
  WT[((size_t)mat * H_ + c) * kdim + kk] = W[i];
}

__global__ void transpose_head_kernel(const float* __restrict__ kw, const float* __restrict__ sw,
                                      const float* __restrict__ vw, float* __restrict__ kT,
                                      float* __restrict__ sT, float* __restrict__ vT) {
  // [K][H][D] -> [K*D][H] with outcol = k*DH + d
  const int i = blockIdx.x * 256 + threadIdx.x;
  if (i >= KH * H_ * DH) return;
  const int d = i % DH, h = (i / DH) % H_, k = i / (DH * H_);
  const size_t o = (size_t)(k * DH + d) * H_ + h;
  kT[o] = kw[i]; sT[o] = sw[i]; vT[o] = vw[i];
}

// ---------------- BatchNorm statistics ----------------

__global__ void zero_ws_kernel(float* __restrict__ p, int n) {
  const int i = blockIdx.x * blockDim.x + threadIdx.x;
  if (i < n) p[i] = 0.f;
}

__global__ __launch_bounds__(160) void bn_stats_kernel(const float* __restrict__ states,
                                                       const float* __restrict__ actions,
                                                       float* __restrict__ stats) {
  const int a = blockIdx.y;
  const int b0 = blockIdx.x * 128;
  const int f = threadIdx.x;   // 0..159
  float s = 0.f, s2 = 0.f;
  if (f < IN_) {
    const float* p = states + (size_t)a * B_N * IN_ + f;
    for (int r = 0; r < 128; ++r) { float x = p[(size_t)(b0 + r) * IN_];  s += x; s2 += x * x; }
  } else {
    const float* p = actions + (size_t)a * B_N * OUT_ + (f - IN_);
    for (int r = 0; r < 128; ++r) { float x = p[(size_t)(b0 + r) * OUT_]; s += x; s2 += x * x; }
  }
  atomicAdd(&stats[(a * SAF + f) * 2 + 0], s);
  atomicAdd(&stats[(a * SAF + f) * 2 + 1], s2);
}

__global__ void bn_finalize_kernel(const float* __restrict__ stats, float* __restrict__ meanstd) {
  const int i = blockIdx.x * blockDim.x + threadIdx.x;
  if (i >= A_ * SAF) return;
  const float invB = 1.f / (float)B_N;
  const float mu  = stats[i * 2] * invB;
  const float var = stats[i * 2 + 1] * invB - mu * mu;
  meanstd[i * 2]     = mu;
  meanstd[i * 2 + 1] = rsqrtf(var + EPS_);
}

// ---------------- Fused encoder ----------------

__global__ __launch_bounds__(256) void
encoder_kernel(const float* __restrict__ states, const float* __restrict__ actions,
               const float* __restrict__ meanstd,
               const float* __restrict__ encT, const float* __restrict__ enc_b,
               const float* __restrict__ aencT, const float* __restrict__ aenc_b,
               const float* __restrict__ keyT, const float* __restrict__ selT,
               const float* __restrict__ valT, const float* __restrict__ val_b,
               float* __restrict__ sa_enc_g, float* __restrict__ keys_g,
               float* __restrict__ vals_g, float* __restrict__ sels_g) {
  __shared__ float ldsS[BT * 132];        // TDM-staged states tile (row stride 128 + 4 pads)
  __shared__ float ldsA[BT * 33];         // TDM-staged actions tile (row stride 32 + 1 pad)
  __shared__ float ldsE[BT * (H_ + 1)];   // a_enc, then sa_enc

  const int a = blockIdx.y;
  const int b0 = blockIdx.x * BT;
  const int tid = threadIdx.x;
  const int lane = tid & 31;
  const int wv = tid >> 5;
  const int m = lane & 15;
  const int kh = lane >> 4;

  // async DMA the raw input tiles into LDS (wave 0 issues; TDM ignores EXEC)
  if (tid < 32) {
    tdm_load((unsigned)(uintptr_t)ldsS, states + ((size_t)a * B_N + b0) * IN_,
             IN_, B_N, IN_, BT, 0, IN_, 0, 0);
    tdm_load((unsigned)(uintptr_t)ldsA, actions + ((size_t)a * B_N + b0) * OUT_,
             OUT_, B_N, OUT_, BT, 0, OUT_, 0, 0);
    __builtin_amdgcn_s_wait_tensorcnt(0);
  }
  __syncthreads();

  // batch-normalize in place
  for (int idx = tid; idx < BT * SAF; idx += 256) {
    const int r = idx / SAF, f = idx % SAF;
    const float mu = meanstd[(a * SAF + f) * 2 + 0];
    const float is = meanstd[(a * SAF + f) * 2 + 1];
    if (f < IN_) { float* p = &ldsS[r * 132 + padidx(f)]; *p = (*p - mu) * is; }
    else         { float* p = &ldsA[r * 33 + (f - IN_)];  *p = (*p - mu) * is; }
  }
  __syncthreads();

  auto xA  = [&](int r, int kk) -> float { return ldsA[r * 33 + kk]; };
  auto xSA = [&](int r, int kk) -> float {
    return kk < IN_ ? ldsS[r * 132 + padidx(kk)] : ldsA[r * 33 + (kk - IN_)];
  };
  auto xE  = [&](int r, int kk) -> float { return ldsE[r * (H_ + 1) + kk]; };

  const size_t gb = (size_t)a * B_N + b0;
  v8f acc[4];

  // a_enc = lrelu(bn(actions) @ aenc_w + aenc_b), K = 32
  zero4(acc);
  gemm_col(xA, OUT_, aencT + (size_t)a * H_ * OUT_, OUT_, wv, m, kh, acc);
  store_col(acc, aenc_b + a * H_, true, nullptr, 0, ldsE, H_ + 1, wv, m, kh);
  __syncthreads();

  // keys = a_enc @ key_w
  zero4(acc);
  gemm_col(xE, H_, keyT, H_, wv, m, kh, acc);
  store_col(acc, nullptr, false, keys_g, gb, nullptr, 0, wv, m, kh);

  // vals = lrelu(a_enc @ val_w + val_b)   (val_b flat [K*D] == per-col bias)
  zero4(acc);
  gemm_col(xE, H_, valT, H_, wv, m, kh, acc);
  store_col(acc, val_b, true, vals_g, gb, nullptr, 0, wv, m, kh);
  __syncthreads();   // done reading a_enc

  // sa_enc = lrelu(bn(sa) @ enc_w + enc_b), K = 160
  zero4(acc);
  gemm_col(xSA, SAF, encT + (size_t)a * H_ * SAF, SAF, wv, m, kh, acc);
  store_col(acc, enc_b + a * H_, true, sa_enc_g, gb, ldsE, H_ + 1, wv, m, kh);
  __syncthreads();

  // sels = sa_enc @ sel_w
  zero4(acc);
  gemm_col(xE, H_, selT, H_, wv, m, kh, acc);
  store_col(acc, nullptr, false, sels_g, gb, nullptr, 0, wv, m, kh);
}

// ---------------- Attention: masked 8-agent softmax per (b, head) ----------------

__global__ __launch_bounds__(256) void
attn_kernel(const float* __restrict__ keys_g, const float* __restrict__ vals_g,
            const float* __restrict__ sels_g, float* __restrict__ other_g) {
  __shared__ float ldsK[A_ * ATT_BT * 132];
  __shared__ float ldsV[A_ * ATT_BT * 132];
  const int b0 = blockIdx.x * ATT_BT;
  const int tid = threadIdx.x;

  // 3D TDM tile: dim0 = 128 cols, dim1 = 8 batch rows, dim2 = 8 agents (stride B*H)
  if (tid < 32) {
    tdm_load((unsigned)(uintptr_t)ldsK, keys_g + (size_t)b0 * H_,
             H_, B_N, H_, ATT_BT, A_, H_, (unsigned long long)B_N * H_, A_);
    tdm_load((unsigned)(uintptr_t)ldsV, vals_g + (size_t)b0 * H_,
             H_, B_N, H_, ATT_BT, A_, H_, (unsigned long long)B_N * H_, A_);
    __builtin_amdgcn_s_wait_tensorcnt(0);
  }
  __syncthreads();

  const int bl = tid & (ATT_BT - 1);
  const int i  = (tid >> 3) & 7;   // query agent
  const int k  = tid >> 6;         // head
  const int b  = b0 + bl;
  const int lbase = k * 33;        // padded offset of head column block (k*32 + k)

  float sel[DH];
  const size_t sbase = ((size_t)i * B_N + b) * H_ + k * DH;
#pragma unroll
  for (int d = 0; d < DH; ++d) sel[d] = sels_g[sbase + d];

  const float scale = 0.17677669529663687f;  // 1/sqrt(D)
  float logits[A_];
#pragma unroll
  for (int j = 0; j < A_; ++j) {
    const float* kr = &ldsK[(j * ATT_BT + bl) * 132 + lbase];
    float dot = 0.f;
#pragma unroll
    for (int d = 0; d < DH; ++d) dot += sel[d] * kr[d];
    logits[j] = dot * scale;
  }
  float mx = -1e30f;
#pragma unroll
  for (int j = 0; j < A_; ++j) if (j != i && logits[j] > mx) mx = logits[j];
  float p[A_]; float sum = 0.f;
#pragma unroll
  for (int j = 0; j < A_; ++j) {
    p[j] = (j == i) ? 0.f : __expf(logits[j] - mx);   // self masked to -inf
    sum += p[j];
  }
  const float inv = 1.f / sum;
  float accv[DH];
#pragma unroll
  for (int d = 0; d < DH; ++d) accv[d] = 0.f;
#pragma unroll
  for (int j = 0; j < A_; ++j) {
    const float* vr = &ldsV[(j * ATT_BT + bl) * 132 + lbase];
    const float pj = p[j] * inv;
#pragma unroll
    for (int d = 0; d < DH; ++d) accv[d] += pj * vr[d];
  }
  const size_t obase = ((size_t)i * B_N + b) * H_ + k * DH;
#pragma unroll
  for (int d = 0; d < DH; ++d) other_g[obase + d] = accv[d];
}

// ---------------- Critic ----------------

__global__ __launch_bounds__(256) void
critic_kernel(const float* __restrict__ sa_enc_g, const float* __restrict__ other_g,
              const float* __restrict__ c1T, const float* __restrict__ c1_b,
              const float* __restrict__ c2_w, const float* __restrict__ c2_b,
              float* __restrict__ qout) {
  __shared__ float ldsS2[BT * 132];       // sa_enc tile
  __shared__ float ldsO[BT * 132];        // attended tile
  __shared__ float ldsH[BT * (H_ + 1)];   // hidden
  const int a = blockIdx.y;
  const int b0 = blockIdx.x * BT;
  const int tid = threadIdx.x;
  const int lane = tid & 31, wv = tid >> 5, m = lane & 15, kh = lane >> 4;

  if (tid < 32) {
    tdm_load((unsigned)(uintptr_t)ldsS2, sa_enc_g + ((size_t)a * B_N + b0) * H_,
             H_, B_N, H_, BT, 0, H_, 0, 0);
    tdm_load((unsigned)(uintptr_t)ldsO, other_g + ((size_t)a * B_N + b0) * H_,
             H_, B_N, H_, BT, 0, H_, 0, 0);
    __builtin_amdgcn_s_wait_tensorcnt(0);
  }
  __syncthreads();

  auto xC = [&](int r, int kk) -> float {
    return kk < H_ ? ldsS2[r * 132 + padidx(kk)] : ldsO[r * 132 + padidx(kk - H_)];
  };

  v8f acc[4];
  zero4(acc);
  gemm_col(xC, 2 * H_, c1T + (size_t)a * H_ * 2 * H_, 2 * H_, wv, m, kh, acc);
  store_col(acc, c1_b + a * H_, true, nullptr, 0, ldsH, H_ + 1, wv, m, kh);
  __syncthreads();

  if (tid < BT) {
    const float* hr = &ldsH[tid * (H_ + 1)];
    const float* w  = c2_w + a * H_;
    float dot = 0.f;
    for (int c = 0; c < H_; ++c) dot += hr[c] * w[c];
    qout[(size_t)a * B_N + b0 + tid] = dot + c2_b[a];
  }
}

// ---------------- launch ----------------

extern "C" void kernel_launch(void* const* d_in, const int* in_sizes, int n_in,
                              void* d_out, int out_size, void* d_ws, size_t ws_size,
                              hipStream_t stream) {
  const float* states  = (const float*)d_in[0];
  const float* actions = (const float*)d_in[1];
  const float* enc_w   = (const float*)d_in[2];
  const float* enc_b   = (const float*)d_in[3];
  const float* aenc_w  = (const float*)d_in[4];
  const float* aenc_b  = (const float*)d_in[5];
  const float* key_w   = (const float*)d_in[6];
  const float* sel_w   = (const float*)d_in[7];
  const float* val_w   = (const float*)d_in[8];
  const float* val_b   = (const float*)d_in[9];
  const float* c1_w    = (const float*)d_in[10];
  const float* c1_b    = (const float*)d_in[11];
  const float* c2_w    = (const float*)d_in[12];
  const float* c2_b    = (const float*)d_in[13];

  float* ws = (float*)d_ws;
  float* stats   = ws;                               // A*SAF*2 running sums
  float* meanstd = ws + (size_t)A_ * SAF * 2;        // A*SAF*2 mean/invstd
  float* encT    = ws + 8192;                        // [A][H][SAF]
  float* aencT   = encT  + (size_t)A_ * H_ * SAF;    // [A][H][OUT]
  float* keyT    = aencT + (size_t)A_ * H_ * OUT_;   // [K*D][H]
  float* selT    = keyT  + (size_t)KH * DH * H_;
  float* valT    = selT  + (size_t)KH * DH * H_;
  float* c1T     = valT  + (size_t)KH * DH * H_;     // [A][H][2H]
  float* sa_enc  = ws + (1 << 19);                   // big activations from 2 MB mark
  const size_t N = (size_t)A_ * B_N * H_;
  float* keys  = sa_enc + N;
  float* vals  = keys + N;
  float* sels  = vals + N;
  float* other = sels + N;

  zero_ws_kernel<<<(A_ * SAF * 2 + 255) / 256, 256, 0, stream>>>(stats, A_ * SAF * 2);
  transpose_std_kernel<<<(A_ * SAF * H_ + 255) / 256, 256, 0, stream>>>(enc_w, encT, A_, SAF);
  transpose_std_kernel<<<(A_ * OUT_ * H_ + 255) / 256, 256, 0, stream>>>(aenc_w, aencT, A_, OUT_);
  transpose_std_kernel<<<(A_ * 2 * H_ * H_ + 255) / 256, 256, 0, stream>>>(c1_w, c1T, A_, 2 * H_);
  transpose_head_kernel<<<(KH * H_ * DH + 255) / 256, 256, 0, stream>>>(
      key_w, sel_w, val_w, keyT, selT, valT);
  bn_stats_kernel<<<dim3(B_N / 128, A_), 160, 0, stream>>>(states, actions, stats);
  bn_finalize_kernel<<<(A_ * SAF + 255) / 256, 256, 0, stream>>>(stats, meanstd);
  encoder_kernel<<<dim3(B_N / BT, A_), 256, 0, stream>>>(
      states, actions, meanstd, encT, enc_b, aencT, aenc_b,
      keyT, selT, valT, val_b, sa_enc, keys, vals, sels);
  attn_kernel<<<B_N / ATT_BT, 256, 0, stream>>>(keys, vals, sels, other);
  critic_kernel<<<dim3(B_N / BT, A_), 256, 0, stream>>>(
      sa_enc, other, c1T, c1_b, c2_w, c2_b, (float*)d_out);
}